// Mamba1_11699490914657
// MI455X (gfx1250) — compile-verified
//
#include <hip/hip_runtime.h>
#include <hip/hip_bf16.h>
#include <math.h>

typedef __attribute__((ext_vector_type(2))) float v2f;
typedef __attribute__((ext_vector_type(8))) float v8f;
typedef int v4i_vs __attribute__((vector_size(16)));   // matches builtin's pointee type

#define D_MODEL 1024
#define D_STATE 16
#define D_INNER 2048
#define DT_RANK 64
#define D_CONV  4
#define B_SZ    2
#define SEQ_L   2048
#define M_TOT   (B_SZ * SEQ_L)   // 4096 token rows

// CDNA5 async global->LDS copy (ASYNCcnt-tracked, bypasses VGPRs)
#if defined(__has_builtin)
#if __has_builtin(__builtin_amdgcn_global_load_async_to_lds_b128)
#define USE_ASYNC_LDS 1
#endif
#endif

#define AS_GLOBAL __attribute__((address_space(1)))
#define AS_LOCAL  __attribute__((address_space(3)))

__device__ __forceinline__ void wait_async0() {
#if defined(__has_builtin) && __has_builtin(__builtin_amdgcn_s_wait_asynccnt)
    __builtin_amdgcn_s_wait_asynccnt(0);
#else
    asm volatile("s_wait_asynccnt 0x0" ::: "memory");
#endif
}

__device__ __forceinline__ float silu_f(float x) { return x / (1.f + __expf(-x)); }

// ---------------------------------------------------------------------------
// fp32 WMMA GEMM:  C[M,N] = A[M,K] * W[N,K]^T   (+ optional bias/softplus)
//   128 threads (4 waves), 64x64 block tile, each wave a 32x32 region via
//   2x2 V_WMMA_F32_16X16X4_F32 accumulators.  K staged in 16-wide LDS panels,
//   DOUBLE-BUFFERED with async global->LDS loads: panel k+1 is in flight
//   while panel k feeds the WMMAs; only s_wait_asynccnt+barrier at boundaries.
//   EPI == 0 : plain store;  EPI == 1 : v = softplus(v + bias[col])
// ---------------------------------------------------------------------------
template <int EPI>
__global__ __launch_bounds__(128)
void wmma_gemm_kernel(const float* __restrict__ A, int lda,
                      const float* __restrict__ W, int ldw,
                      const float* __restrict__ bias,
                      float* __restrict__ C, int ldc,
                      int M, int N, int K)
{
    // row pad 20 floats: rows stay 16B-aligned and 20*r mod 64 is a
    // permutation over 16 rows -> conflict-free ds reads of fragments
    __shared__ float As[2][64][20];
    __shared__ float Ws[2][64][20];

    const int tid  = threadIdx.x;
    const int lane = tid & 31;
    const int wave = tid >> 5;
    const int tm   = (wave >> 1) * 32;
    const int tn   = (wave & 1) * 32;
    const int m0   = blockIdx.y * 64;
    const int n0   = blockIdx.x * 64;

    v8f acc[2][2] = {};

    const int mr   = lane & 15;
    const int half = lane >> 4;

    // stage one 64x16 panel pair (A and W) into LDS buffer `buf`
    auto stage = [&](int buf, int k0) {
        #pragma unroll
        for (int s = 0; s < 2; ++s) {
            const int fid = tid + s * 128;          // 256 float4 slots/panel
            const int row = fid >> 2;
            const int c4  = (fid & 3) * 4;
            const float* gA = &A[(size_t)(m0 + row) * lda + k0 + c4];
#ifdef USE_ASYNC_LDS
            __builtin_amdgcn_global_load_async_to_lds_b128(
                (AS_GLOBAL v4i_vs*)gA, (AS_LOCAL v4i_vs*)&As[buf][row][c4], 0, 0);
#else
            *(float4*)&As[buf][row][c4] = *(const float4*)gA;
#endif
            const int nrow = n0 + row;
            if (nrow < N) {
                const float* gW = &W[(size_t)nrow * ldw + k0 + c4];
#ifdef USE_ASYNC_LDS
                __builtin_amdgcn_global_load_async_to_lds_b128(
                    (AS_GLOBAL v4i_vs*)gW, (AS_LOCAL v4i_vs*)&Ws[buf][row][c4], 0, 0);
#else
                *(float4*)&Ws[buf][row][c4] = *(const float4*)gW;
#endif
            } else {
                *(float4*)&Ws[buf][row][c4] = make_float4(0.f, 0.f, 0.f, 0.f);
            }
        }
    };

    stage(0, 0);
    wait_async0();
    __syncthreads();

    const int nk = K >> 4;
    for (int ki = 0; ki < nk; ++ki) {
        const int cur = ki & 1;
        if (ki + 1 < nk) stage(cur ^ 1, (ki + 1) << 4);   // prefetch next panel

        #pragma unroll
        for (int kk = 0; kk < 4; ++kk) {
            const int ks = kk * 4 + half * 2;
            v2f a0 = { As[cur][tm + mr][ks],      As[cur][tm + mr][ks + 1] };
            v2f a1 = { As[cur][tm + 16 + mr][ks], As[cur][tm + 16 + mr][ks + 1] };
            v2f b0 = { Ws[cur][tn + mr][ks],      Ws[cur][tn + mr][ks + 1] };
            v2f b1 = { Ws[cur][tn + 16 + mr][ks], Ws[cur][tn + 16 + mr][ks + 1] };

            acc[0][0] = __builtin_amdgcn_wmma_f32_16x16x4_f32(false, a0, false, b0, (short)0, acc[0][0], false, false);
            acc[0][1] = __builtin_amdgcn_wmma_f32_16x16x4_f32(false, a0, false, b1, (short)0, acc[0][1], false, false);
            acc[1][0] = __builtin_amdgcn_wmma_f32_16x16x4_f32(false, a1, false, b0, (short)0, acc[1][0], false, false);
            acc[1][1] = __builtin_amdgcn_wmma_f32_16x16x4_f32(false, a1, false, b1, (short)0, acc[1][1], false, false);
        }

        wait_async0();        // this wave's async writes into buffer cur^1 landed
        __syncthreads();      // all waves' panel writes visible
    }

    // epilogue: v8f element r -> row r + 8*half, col = lane&15
    const int nc = lane & 15;
    #pragma unroll
    for (int i = 0; i < 2; ++i) {
        #pragma unroll
        for (int j = 0; j < 2; ++j) {
            const int rbase = m0 + tm + 16 * i + 8 * half;
            const int col   = n0 + tn + 16 * j + nc;
            if (col < N) {
                const float bv = (EPI == 1) ? bias[col] : 0.f;
                #pragma unroll
                for (int r = 0; r < 8; ++r) {
                    float v = acc[i][j][r];
                    if (EPI == 1) {
                        v += bv;
                        v = (v > 20.f) ? v : log1pf(__expf(v));  // softplus
                    }
                    C[(size_t)(rbase + r) * ldc + col] = v;
                }
            }
        }
    }
}

// ---------------------------------------------------------------------------
// Causal depthwise conv1d (k=4) + bias + SiLU.  Coalesced over d.
// ---------------------------------------------------------------------------
__global__ __launch_bounds__(256)
void conv_silu_kernel(const float* __restrict__ xr,
                      const float* __restrict__ cw,   // (D_INNER, 1, 4)
                      const float* __restrict__ cb,   // (D_INNER)
                      float* __restrict__ xc)         // (M_TOT, D_INNER)
{
    const int idx = blockIdx.x * 256 + threadIdx.x;
    if (idx >= M_TOT * D_INNER) return;
    const int d  = idx % D_INNER;
    const int ml = idx / D_INNER;
    const int l  = ml % SEQ_L;
    const int b  = ml / SEQ_L;

    float s = cb[d];
    #pragma unroll
    for (int k = 0; k < D_CONV; ++k) {
        const int ll = l - (D_CONV - 1) + k;
        if (ll >= 0)
            s += cw[d * D_CONV + k] * xr[(size_t)(b * SEQ_L + ll) * (2 * D_INNER) + d];
    }
    xc[idx] = silu_f(s);
}

// ---------------------------------------------------------------------------
// Selective scan fused with (+u*D)*silu(res) gating.  One thread per (b,d)
// channel; 16-state recurrence in registers; per-step B/C staged via LDS in
// 64-step chunks (broadcast reads).
// ---------------------------------------------------------------------------
__global__ __launch_bounds__(128)
void scan_kernel(const float* __restrict__ xdbl,   // (M_TOT, 96)
                 const float* __restrict__ delta,  // (M_TOT, D_INNER)
                 const float* __restrict__ xc,     // (M_TOT, D_INNER)
                 const float* __restrict__ xr,     // (M_TOT, 2*D_INNER)
                 const float* __restrict__ A_log,  // (D_INNER, D_STATE)
                 const float* __restrict__ Dp,     // (D_INNER)
                 float* __restrict__ yg)           // (M_TOT, D_INNER)
{
    const int b = blockIdx.x / (D_INNER / 128);
    const int d = (blockIdx.x % (D_INNER / 128)) * 128 + threadIdx.x;

    float Ad[D_STATE];
    #pragma unroll
    for (int n = 0; n < D_STATE; ++n)
        Ad[n] = -__expf(A_log[d * D_STATE + n]);
    const float Dd = Dp[d];

    float st[D_STATE] = {};

    __shared__ float sB[64][D_STATE];
    __shared__ float sC[64][D_STATE];

    for (int c0 = 0; c0 < SEQ_L; c0 += 64) {
        for (int i = threadIdx.x; i < 64 * 2 * D_STATE; i += 128) {
            const int t = i >> 5, j = i & 31;
            const float v = xdbl[(size_t)(b * SEQ_L + c0 + t) * 96 + DT_RANK + j];
            if (j < D_STATE) sB[t][j] = v;
            else             sC[t][j - D_STATE] = v;
        }
        __syncthreads();

        for (int t = 0; t < 64; ++t) {
            const size_t ml  = (size_t)b * SEQ_L + c0 + t;
            const size_t idx = ml * D_INNER + d;
            const float dt = delta[idx];
            const float u  = xc[idx];
            const float du = dt * u;
            float y = 0.f;
            #pragma unroll
            for (int n = 0; n < D_STATE; ++n) {
                st[n] = __expf(dt * Ad[n]) * st[n] + du * sB[t][n];
                y += st[n] * sC[t][n];
            }
            const float r = xr[ml * (2 * D_INNER) + D_INNER + d];
            yg[idx] = (y + u * Dd) * silu_f(r);
        }
        __syncthreads();
    }
}

// ---------------------------------------------------------------------------
extern "C" void kernel_launch(void* const* d_in, const int* in_sizes, int n_in,
                              void* d_out, int out_size, void* d_ws, size_t ws_size,
                              hipStream_t stream)
{
    (void)in_sizes; (void)n_in; (void)out_size; (void)ws_size;
    const float* x     = (const float*)d_in[0];  // (B,L,1024)
    const float* inw   = (const float*)d_in[1];  // (4096,1024)
    const float* convw = (const float*)d_in[2];  // (2048,1,4)
    const float* convb = (const float*)d_in[3];  // (2048)
    const float* xpw   = (const float*)d_in[4];  // (96,2048)
    const float* dtw   = (const float*)d_in[5];  // (2048,64)
    const float* dtb   = (const float*)d_in[6];  // (2048)
    const float* alog  = (const float*)d_in[7];  // (2048,16)
    const float* Dp    = (const float*)d_in[8];  // (2048)
    const float* outw  = (const float*)d_in[9];  // (1024,2048)
    float* out = (float*)d_out;

    float* ws   = (float*)d_ws;
    float* xr   = ws;                     size_t o = (size_t)M_TOT * 2 * D_INNER; // 64 MB
    float* xc   = ws + o;                 o += (size_t)M_TOT * D_INNER;           // 32 MB
    float* xdbl = ws + o;                 o += (size_t)M_TOT * 96;                // 1.5 MB
    float* dlt  = ws + o;                 o += (size_t)M_TOT * D_INNER;           // 32 MB
    float* yg   = ws + o;                                                         // 32 MB

    const dim3 blk(128);

    // 1) in_proj -> x_and_res
    wmma_gemm_kernel<0><<<dim3((2 * D_INNER) / 64, M_TOT / 64), blk, 0, stream>>>(
        x, D_MODEL, inw, D_MODEL, nullptr, xr, 2 * D_INNER, M_TOT, 2 * D_INNER, D_MODEL);

    // 2) causal depthwise conv + SiLU -> xc
    conv_silu_kernel<<<(M_TOT * D_INNER + 255) / 256, 256, 0, stream>>>(xr, convw, convb, xc);

    // 3) x_proj -> x_dbl
    wmma_gemm_kernel<0><<<dim3(2, M_TOT / 64), blk, 0, stream>>>(
        xc, D_INNER, xpw, D_INNER, nullptr, xdbl, 96, M_TOT, 96, D_INNER);

    // 4) dt_proj + softplus -> delta
    wmma_gemm_kernel<1><<<dim3(D_INNER / 64, M_TOT / 64), blk, 0, stream>>>(
        xdbl, 96, dtw, DT_RANK, dtb, dlt, D_INNER, M_TOT, D_INNER, DT_RANK);

    // 5) selective scan + gating -> yg
    scan_kernel<<<B_SZ * (D_INNER / 128), 128, 0, stream>>>(xdbl, dlt, xc, xr, alog, Dp, yg);

    // 6) out_proj -> out
    wmma_gemm_kernel<0><<<dim3(D_MODEL / 64, M_TOT / 64), blk, 0, stream>>>(
        yg, D_INNER, outw, D_INNER, nullptr, out, D_MODEL, M_TOT, D_MODEL, D_INNER);
}